// S4Block_29901562315293
// MI455X (gfx1250) — compile-verified
//
#include <hip/hip_runtime.h>
#include <math.h>

typedef __attribute__((ext_vector_type(2))) float v2f;
typedef __attribute__((ext_vector_type(4))) float v4f;
typedef __attribute__((ext_vector_type(8))) float v8f;

#define H 256
#define N2 32
#define BATCH 16
#define LSEQ 4096
#define MROWS (BATCH * LSEQ)   // 65536

__device__ __forceinline__ float wave_reduce_sum(float v) {
    #pragma unroll
    for (int off = 16; off > 0; off >>= 1) v += __shfl_xor(v, off, 32);
    return v;
}

__device__ __forceinline__ float gelu_exact(float x) {
    return 0.5f * x * (1.0f + erff(x * 0.70710678118654752f));
}

// ---------------------------------------------------------------------------
// K0: fold ZOH discretization into per-mode params.
// P layout: [wr | wi | cfr | cfi | cbr | cbi], each H*N2 floats.
// w = exp(dt*A);  Ctil = 2 * C * (w - 1) / A
// ---------------------------------------------------------------------------
__global__ void param_kernel(const float* __restrict__ log_dt,
                             const float* __restrict__ A_re,
                             const float* __restrict__ A_im,
                             const float* __restrict__ C_re,
                             const float* __restrict__ C_im,
                             float* __restrict__ P) {
    int idx = blockIdx.x * blockDim.x + threadIdx.x;
    const int HN = H * N2;
    if (idx >= HN) return;
    int h = idx / N2;
    float dt = expf(log_dt[h]);
    float ar = A_re[idx], ai = A_im[idx];
    float er = expf(dt * ar);
    float wr = er * cosf(dt * ai);
    float wi = er * sinf(dt * ai);
    float inv = 1.0f / (ar * ar + ai * ai);
    float fr = ((wr - 1.0f) * ar + wi * ai) * inv;   // (w-1)/A via conj
    float fi = (wi * ar - (wr - 1.0f) * ai) * inv;
    float cr = C_re[idx], ci = C_im[idx];            // c=0 (forward)
    P[2 * HN + idx] = 2.0f * (cr * fr - ci * fi);
    P[3 * HN + idx] = 2.0f * (cr * fi + ci * fr);
    cr = C_re[HN + idx]; ci = C_im[HN + idx];        // c=1 (backward)
    P[4 * HN + idx] = 2.0f * (cr * fr - ci * fi);
    P[5 * HN + idx] = 2.0f * (cr * fi + ci * fr);
    P[idx] = wr;
    P[HN + idx] = wi;
}

// ---------------------------------------------------------------------------
// Repack weights (K x N) into row-pair-interleaved form:
// Wp[(p*N + n)*2 + {0,1}] = W[(2p)*N + n], W[(2p+1)*N + n]
// so a WMMA B-fragment is one coalesced b64 load.
// ---------------------------------------------------------------------------
__global__ void repack_kernel(const float* __restrict__ W, float* __restrict__ Wp,
                              int K, int N) {
    int idx = blockIdx.x * blockDim.x + threadIdx.x;
    int total = (K >> 1) * N;
    if (idx >= total) return;
    int p = idx / N, n = idx % N;
    Wp[(size_t)idx * 2]     = W[(size_t)(2 * p) * N + n];
    Wp[(size_t)idx * 2 + 1] = W[(size_t)(2 * p + 1) * N + n];
}

// ---------------------------------------------------------------------------
// LayerNorm: one wave per row of H=256, vectorized b128 loads/stores.
// ---------------------------------------------------------------------------
__global__ __launch_bounds__(256)
void ln_kernel(const float* __restrict__ X, const float* __restrict__ g,
               const float* __restrict__ bta, float* __restrict__ Out) {
    int wave = (blockIdx.x * blockDim.x + threadIdx.x) >> 5;
    int lane = threadIdx.x & 31;
    const v4f* row4 = (const v4f*)(X + (size_t)wave * H);
    v4f c0 = row4[lane];
    v4f c1 = row4[lane + 32];
    float s = (c0.x + c0.y) + (c0.z + c0.w) + (c1.x + c1.y) + (c1.z + c1.w);
    s = wave_reduce_sum(s);
    float mean = s * (1.0f / H);
    v4f d0 = c0 - mean, d1 = c1 - mean;
    float q = (d0.x * d0.x + d0.y * d0.y) + (d0.z * d0.z + d0.w * d0.w)
            + (d1.x * d1.x + d1.y * d1.y) + (d1.z * d1.z + d1.w * d1.w);
    q = wave_reduce_sum(q);
    float rstd = rsqrtf(q * (1.0f / H) + 1e-5f);
    const v4f* g4 = (const v4f*)g;
    const v4f* b4 = (const v4f*)bta;
    v4f* o4 = (v4f*)(Out + (size_t)wave * H);
    o4[lane]      = d0 * rstd * g4[lane]      + b4[lane];
    o4[lane + 32] = d1 * rstd * g4[lane + 32] + b4[lane + 32];
}

// ---------------------------------------------------------------------------
// K2: bidirectional diagonal-SSM scan + D-skip + GELU.
// Block = 1024 threads = 32 waves; wave w owns channel h = h0 + w; lane = mode.
// Chunk-of-32 structure: inner loop fully unrolled so the 32 butterfly
// reduction trees interleave for ILP; flushes are unconditional.
// Yg may alias U (block-disjoint 128B-aligned h-columns; row t written only
// after all block waves finished reading u[>=t]).
// ---------------------------------------------------------------------------
__global__ __launch_bounds__(1024)
void scan_kernel(const float* __restrict__ U,      // (B,L,H)
                 const float* __restrict__ P,
                 const float* __restrict__ Dskip,
                 float* __restrict__ Yfwd,         // (B,H,L) scratch
                 float* __restrict__ Yg) {         // (B,L,H) post-GELU
    __shared__ float tile[32 * 33];
    int b  = blockIdx.x >> 3;             // H/32 == 8 blocks per batch
    int h0 = (blockIdx.x & 7) * 32;
    int w    = threadIdx.x >> 5;
    int lane = threadIdx.x & 31;
    int h = h0 + w;
    const int HN = H * N2;
    int pidx = h * N2 + lane;
    float wr  = P[pidx],          wi  = P[HN + pidx];
    float cfr = P[2 * HN + pidx], cfi = P[3 * HN + pidx];
    float cbr = P[4 * HN + pidx], cbi = P[5 * HN + pidx];
    float Dh = Dskip[h];
    const float* ub = U + (size_t)b * LSEQ * H + h;       // stride H over t
    float* yf = Yfwd + ((size_t)b * H + h) * LSEQ;

    // ---- forward scan ----
    float sr = 0.0f, si = 0.0f;
    for (int t0 = 0; t0 < LSEQ; t0 += 32) {
        float acc = 0.0f;
        #pragma unroll
        for (int j = 0; j < 32; ++j) {
            float u = ub[(size_t)(t0 + j) * H];           // lane-uniform
            float nsr = sr * wr - si * wi + u;
            float nsi = sr * wi + si * wr;
            sr = nsr; si = nsi;
            float yv = wave_reduce_sum(cfr * sr - cfi * si);
            if (lane == j) acc = yv;
        }
        yf[t0 + lane] = acc;                              // coalesced 128B
    }

    // ---- backward scan + combine + GELU + transpose store ----
    float rr = 0.0f, ri = 0.0f;
    for (int t0 = LSEQ - 32; t0 >= 0; t0 -= 32) {
        float acc = 0.0f;
        #pragma unroll
        for (int j = 31; j >= 0; --j) {
            int t = t0 + j;
            float ybwd = wave_reduce_sum(cbr * rr - cbi * ri);  // uses r[t]
            float u = ub[(size_t)t * H];
            float val = gelu_exact(yf[t] + ybwd + Dh * u);
            if (lane == j) acc = val;
            float nrr = rr * wr - ri * wi + u;            // r[t-1] = u[t] + w*r[t]
            float nri = rr * wi + ri * wr;
            rr = nrr; ri = nri;
        }
        tile[lane * 33 + w] = acc;                        // lane = t_local
        __syncthreads();
        Yg[((size_t)b * LSEQ + (t0 + w)) * H + h0 + lane] = tile[w * 33 + lane];
        __syncthreads();
    }
}

// ---------------------------------------------------------------------------
// fp32 WMMA GEMM tiles: one wave per pair of 16x16 output tiles, K step 4.
// A frag (16x4): lane<16 -> row m0+lane cols {k,k+1}; lane>=16 -> {k+2,k+3}
// B frag from repacked weights: single b64 load per frag.
// C/D: VGPR r -> rows {m0+r | m0+r+8}, lanes = cols.
// ---------------------------------------------------------------------------

// G1: Z = Y @ Wout + bout ; GLU pair (n, n+H) ; out = X + a*sigmoid(b)
__global__ __launch_bounds__(256)
void g1_glu_kernel(const float* __restrict__ Y, const float* __restrict__ Wp,
                   const float* __restrict__ bias, const float* __restrict__ X,
                   float* __restrict__ Out) {
    int wave = (blockIdx.x * blockDim.x + threadIdx.x) >> 5;
    int lane = threadIdx.x & 31;
    int nt = wave & 15;                      // 16 n-tiles over first half H
    int mt = wave >> 4;
    int m0 = mt * 16, n0 = nt * 16;
    int half = lane >> 4, ln = lane & 15;
    const float* arow = Y + (size_t)(m0 + ln) * H;
    const float* bbase = Wp + ((size_t)half * (2 * H) + n0 + ln) * 2;
    v8f acc0 = {}; v8f acc1 = {};
    #pragma unroll 8
    for (int k = 0; k < H; k += 4) {
        v2f a = *(const v2f*)(arow + k + 2 * half);
        const float* bp = bbase + (size_t)(k >> 1) * (2 * H) * 2;
        v2f b0 = *(const v2f*)bp;            // cols n0..  (value half)
        v2f b1 = *(const v2f*)(bp + 2 * H);  // cols n0+H.. (gate half)
        acc0 = __builtin_amdgcn_wmma_f32_16x16x4_f32(false, a, false, b0,
                                                     (short)0, acc0, false, false);
        acc1 = __builtin_amdgcn_wmma_f32_16x16x4_f32(false, a, false, b1,
                                                     (short)0, acc1, false, false);
    }
    float ba = bias[n0 + ln];
    float bb = bias[n0 + ln + H];
    #pragma unroll
    for (int r = 0; r < 8; ++r) {
        int m = m0 + r + 8 * half;
        float av = acc0[r] + ba;
        float bv = acc1[r] + bb;
        float gl = av * (1.0f / (1.0f + expf(-bv)));
        size_t idx = (size_t)m * H + n0 + ln;
        Out[idx] = X[idx] + gl;
    }
}

// G2: T = gelu(h2 @ W2 + b2), N = 2H = 512, K = H; two n-tiles per wave
__global__ __launch_bounds__(256)
void g2_gelu_kernel(const float* __restrict__ A, const float* __restrict__ Wp,
                    const float* __restrict__ bias, float* __restrict__ T) {
    int wave = (blockIdx.x * blockDim.x + threadIdx.x) >> 5;
    int lane = threadIdx.x & 31;
    int nt = wave & 15;                      // tiles (nt, nt+16) over 2H
    int mt = wave >> 4;
    int m0 = mt * 16, n0 = nt * 16;
    int half = lane >> 4, ln = lane & 15;
    const float* arow = A + (size_t)(m0 + ln) * H;
    const float* bbase = Wp + ((size_t)half * (2 * H) + n0 + ln) * 2;
    v8f acc0 = {}; v8f acc1 = {};
    #pragma unroll 8
    for (int k = 0; k < H; k += 4) {
        v2f a = *(const v2f*)(arow + k + 2 * half);
        const float* bp = bbase + (size_t)(k >> 1) * (2 * H) * 2;
        v2f b0 = *(const v2f*)bp;
        v2f b1 = *(const v2f*)(bp + 2 * H);  // cols + 256
        acc0 = __builtin_amdgcn_wmma_f32_16x16x4_f32(false, a, false, b0,
                                                     (short)0, acc0, false, false);
        acc1 = __builtin_amdgcn_wmma_f32_16x16x4_f32(false, a, false, b1,
                                                     (short)0, acc1, false, false);
    }
    float ba = bias[n0 + ln];
    float bb = bias[n0 + ln + H];
    #pragma unroll
    for (int r = 0; r < 8; ++r) {
        int m = m0 + r + 8 * half;
        size_t rowb = (size_t)m * (2 * H) + n0 + ln;
        T[rowb]     = gelu_exact(acc0[r] + ba);
        T[rowb + H] = gelu_exact(acc1[r] + bb);
    }
}

// G3: out = X1 + T @ W3 + b3, K = 2H = 512, N = H; two n-tiles per wave
__global__ __launch_bounds__(256)
void g3_res_kernel(const float* __restrict__ T, const float* __restrict__ Wp,
                   const float* __restrict__ bias, const float* __restrict__ X1,
                   float* __restrict__ Out) {
    int wave = (blockIdx.x * blockDim.x + threadIdx.x) >> 5;
    int lane = threadIdx.x & 31;
    int nt = wave & 7;                       // tiles (nt, nt+8) over H
    int mt = wave >> 3;
    int m0 = mt * 16, n0 = nt * 16;
    int half = lane >> 4, ln = lane & 15;
    const float* arow = T + (size_t)(m0 + ln) * (2 * H);
    const float* bbase = Wp + ((size_t)half * H + n0 + ln) * 2;
    v8f acc0 = {}; v8f acc1 = {};
    #pragma unroll 8
    for (int k = 0; k < 2 * H; k += 4) {
        v2f a = *(const v2f*)(arow + k + 2 * half);
        const float* bp = bbase + (size_t)(k >> 1) * H * 2;
        v2f b0 = *(const v2f*)bp;
        v2f b1 = *(const v2f*)(bp + H);      // cols + 128
        acc0 = __builtin_amdgcn_wmma_f32_16x16x4_f32(false, a, false, b0,
                                                     (short)0, acc0, false, false);
        acc1 = __builtin_amdgcn_wmma_f32_16x16x4_f32(false, a, false, b1,
                                                     (short)0, acc1, false, false);
    }
    float ba = bias[n0 + ln];
    float bb = bias[n0 + ln + H / 2];
    #pragma unroll
    for (int r = 0; r < 8; ++r) {
        int m = m0 + r + 8 * half;
        size_t idx = (size_t)m * H + n0 + ln;
        Out[idx]         = X1[idx]         + acc0[r] + ba;
        Out[idx + H / 2] = X1[idx + H / 2] + acc1[r] + bb;
    }
}

// ---------------------------------------------------------------------------
extern "C" void kernel_launch(void* const* d_in, const int* in_sizes, int n_in,
                              void* d_out, int out_size, void* d_ws, size_t ws_size,
                              hipStream_t stream) {
    const float* x      = (const float*)d_in[0];
    const float* ln1_g  = (const float*)d_in[1];
    const float* ln1_b  = (const float*)d_in[2];
    const float* ln2_g  = (const float*)d_in[3];
    const float* ln2_b  = (const float*)d_in[4];
    const float* log_dt = (const float*)d_in[5];
    const float* A_re   = (const float*)d_in[6];
    const float* A_im   = (const float*)d_in[7];
    const float* C_re   = (const float*)d_in[8];
    const float* C_im   = (const float*)d_in[9];
    const float* Dsk    = (const float*)d_in[10];
    const float* Wout   = (const float*)d_in[11];
    const float* bout   = (const float*)d_in[12];
    const float* W2     = (const float*)d_in[13];
    const float* b2     = (const float*)d_in[14];
    const float* W3     = (const float*)d_in[15];
    const float* b3     = (const float*)d_in[16];
    float* out = (float*)d_out;

    float* ws = (float*)d_ws;
    size_t off = 0;
    float* P   = ws + off; off += (size_t)6 * H * N2;
    float* Wp1 = ws + off; off += (size_t)H * 2 * H;      // repacked Wout
    float* Wp2 = ws + off; off += (size_t)H * 2 * H;      // repacked W2
    float* Wp3 = ws + off; off += (size_t)2 * H * H;      // repacked W3
    off = (off + 255) & ~(size_t)255;
    float* u  = ws + off; off += (size_t)MROWS * H;       // LN1 out; aliased as Yg
    float* yf = ws + off; off += (size_t)MROWS * H;       // fwd scan scratch; aliased as h2
    float* x1 = ws + off; off += (size_t)MROWS * H;       // first residual output
    float* tb = ws + off; off += (size_t)MROWS * 2 * H;   // MLP hidden
    float* yg = u;
    float* h2 = yf;
    (void)ws_size; (void)in_sizes; (void)n_in; (void)out_size;

    // K0: fold discretization + repack weights
    hipLaunchKernelGGL(param_kernel, dim3((H * N2 + 255) / 256), dim3(256), 0, stream,
                       log_dt, A_re, A_im, C_re, C_im, P);
    hipLaunchKernelGGL(repack_kernel, dim3((H / 2 * 2 * H + 255) / 256), dim3(256), 0, stream,
                       Wout, Wp1, H, 2 * H);
    hipLaunchKernelGGL(repack_kernel, dim3((H / 2 * 2 * H + 255) / 256), dim3(256), 0, stream,
                       W2, Wp2, H, 2 * H);
    hipLaunchKernelGGL(repack_kernel, dim3((H * H + 255) / 256), dim3(256), 0, stream,
                       W3, Wp3, 2 * H, H);
    // K1: LN1
    hipLaunchKernelGGL(ln_kernel, dim3(MROWS / 8), dim3(256), 0, stream,
                       x, ln1_g, ln1_b, u);
    // K2: bidirectional scan + D-skip + GELU (writes yg, aliasing u)
    hipLaunchKernelGGL(scan_kernel, dim3(BATCH * (H / 32)), dim3(1024), 0, stream,
                       u, P, Dsk, yf, yg);
    // K3: GLU output projection + residual -> x1
    hipLaunchKernelGGL(g1_glu_kernel, dim3((MROWS / 16) * (H / 16) / 8), dim3(256), 0, stream,
                       yg, Wp1, bout, x, x1);
    // K4: LN2 -> h2
    hipLaunchKernelGGL(ln_kernel, dim3(MROWS / 8), dim3(256), 0, stream,
                       x1, ln2_g, ln2_b, h2);
    // K5: MLP up + GELU -> tb (two n-tiles per wave)
    hipLaunchKernelGGL(g2_gelu_kernel, dim3((MROWS / 16) * (H / 16) / 8), dim3(256), 0, stream,
                       h2, Wp2, b2, tb);
    // K6: MLP down + residual -> out (two n-tiles per wave)
    hipLaunchKernelGGL(g3_res_kernel, dim3((MROWS / 16) * (H / 32) / 8), dim3(256), 0, stream,
                       tb, Wp3, b3, x1, out);
}